// self_attention_46420006535272
// MI455X (gfx1250) — compile-verified
//
#include <hip/hip_runtime.h>

// Self-attention (no-softmax variant): out = Wo @ reinterpret( (q @ k) @ v ) + X
// Rewritten as q @ (k @ v) (softmax output is dead code -> purely linear ->
// matmul associativity). All GEMMs on V_WMMA_F32_16X16X4_F32 (full fp32).
//
// LDS tiles use a pair-interleaved layout: K-rows (2p, 2p+1) stored adjacent
// per column, so each WMMA B-fragment is a single conflict-free ds_load_b64
// (no VGPR repacking before v_wmma).

typedef float v2f __attribute__((ext_vector_type(2)));
typedef float v8f __attribute__((ext_vector_type(8)));

constexpr int CC  = 256;   // C
constexpr int CC2 = 128;   // C/2
constexpr int NN  = 4096;  // T*H*W
constexpr int KCH = 64;    // K rows per LDS chunk
constexpr int PS  = 288;   // floats per K-row-pair (2*128 + 32); 288%64==32 ->
                           // the two lane-halves hit disjoint 32-bank sets
// LDS chunk = (KCH/2)*PS floats = 32*288*4 = 36864 B

__device__ __forceinline__ v8f wmma4(v2f a, v2f b, v8f c) {
  // (neg_a, A, neg_b, B, c_mod, C, reuse_a, reuse_b)
  return __builtin_amdgcn_wmma_f32_16x16x4_f32(false, a, false, b, (short)0, c,
                                               false, false);
}

__device__ __forceinline__ v8f vzero() {
  v8f z = {0.f, 0.f, 0.f, 0.f, 0.f, 0.f, 0.f, 0.f};
  return z;
}

// ---------------------------------------------------------------------------
// Stage 1: q/k/v projections.
//   y[p][b][o][n] = sum_c W_p[o][c]*(X[b][c][n]+pe[c][n]) + b_p[o] + task_p[o][n]
// Block: 768 thr (24 waves) = 3 projections x 8 M-strips, one 128-wide N slab.
// ---------------------------------------------------------------------------
__global__ __launch_bounds__(768) void proj_kernel(
    const float* __restrict__ X,  const float* __restrict__ pe,
    const float* __restrict__ tq, const float* __restrict__ tk,
    const float* __restrict__ tv,
    const float* __restrict__ Wq, const float* __restrict__ bq,
    const float* __restrict__ Wk, const float* __restrict__ bk,
    const float* __restrict__ Wv, const float* __restrict__ bv,
    float* __restrict__ qkv) {
  __shared__ float xs[(KCH / 2) * PS];
  const int b    = blockIdx.y;
  const int n0   = blockIdx.x * 128;
  const int wave = threadIdx.x >> 5;
  const int lane = threadIdx.x & 31;
  const int half = lane >> 4;
  const int lm   = lane & 15;
  const int p    = wave >> 3;        // 0=q 1=k 2=v
  const int m0   = (wave & 7) * 16;  // M-strip

  const float* W    = (p == 0) ? Wq : (p == 1) ? Wk : Wv;
  const float* bias = (p == 0) ? bq : (p == 1) ? bk : bv;
  const float* task = (p == 0) ? tq : (p == 1) ? tk : tv;

  v8f acc[8];
#pragma unroll
  for (int j = 0; j < 8; ++j) acc[j] = vzero();

  for (int kc = 0; kc < CC; kc += KCH) {
    __syncthreads();
    // stage Xp rows [kc, kc+KCH) x cols [n0, n0+128) pair-interleaved
    for (int idx = threadIdx.x; idx < (KCH / 2) * 128; idx += 768) {
      const int pr = idx >> 7, cc = idx & 127;
      const int gi = (kc + 2 * pr) * NN + n0 + cc;
      v2f xv = {X[b * CC * NN + gi] + pe[gi],
                X[b * CC * NN + gi + NN] + pe[gi + NN]};
      *reinterpret_cast<v2f*>(&xs[pr * PS + cc * 2]) = xv;
    }
    __syncthreads();
#pragma unroll 4
    for (int kk = 0; kk < KCH; kk += 4) {
      // A frag: W[m0+lm][kc+kk+2*half + {0,1}] -> one global b64
      const v2f a =
          *reinterpret_cast<const v2f*>(W + (m0 + lm) * CC + kc + kk + 2 * half);
      const float* bbase = &xs[((kk >> 1) + half) * PS + lm * 2];
#pragma unroll
      for (int j = 0; j < 8; ++j) {
        const v2f bf = *reinterpret_cast<const v2f*>(bbase + j * 32);
        acc[j] = wmma4(a, bf, acc[j]);
      }
    }
  }

  float* out = qkv + ((size_t)(p * 4 + b) * CC2) * NN;
#pragma unroll
  for (int j = 0; j < 8; ++j) {
#pragma unroll
    for (int v = 0; v < 8; ++v) {
      const int row = m0 + v + 8 * half;
      const int col = n0 + j * 16 + lm;
      out[(size_t)row * NN + col] = acc[j][v] + bias[row] + task[row * NN + col];
    }
  }
}

// ---------------------------------------------------------------------------
// Stage 2: KV[b][ca][cb] = sum_n k[b][ca][n] * v[b][cb][n]   (128x128, K=4096)
// Grid: B x 8 K-slices (512 each), partials to separate buffers (deterministic,
// no fp atomics). Both fragments are contiguous float2 along K -> b64 loads.
// ---------------------------------------------------------------------------
__global__ __launch_bounds__(256) void kv_kernel(const float* __restrict__ qkv,
                                                 float* __restrict__ kvpart) {
  const int b    = blockIdx.y;
  const int ks   = blockIdx.x;  // K-slice
  const int wave = threadIdx.x >> 5;
  const int lane = threadIdx.x & 31;
  const int half = lane >> 4, lm = lane & 15;
  const int m0   = wave * 16;

  const float* kbuf  = qkv + ((size_t)(4 + b) * CC2) * NN;
  const float* vbuf  = qkv + ((size_t)(8 + b) * CC2) * NN;
  const int    kbase = ks * 512;

  v8f acc[8];
#pragma unroll
  for (int j = 0; j < 8; ++j) acc[j] = vzero();

  const float* ap = kbuf + (m0 + lm) * NN + kbase + 2 * half;
  const float* bp = vbuf + lm * NN + kbase + 2 * half;
#pragma unroll 2
  for (int kk = 0; kk < 512; kk += 4) {
    const v2f a = *reinterpret_cast<const v2f*>(ap + kk);
#pragma unroll
    for (int j = 0; j < 8; ++j) {
      const v2f bf = *reinterpret_cast<const v2f*>(bp + j * 16 * NN + kk);
      acc[j] = wmma4(a, bf, acc[j]);
    }
  }

  float* out = kvpart + ((size_t)(ks * 4 + b) * CC2) * CC2;
#pragma unroll
  for (int j = 0; j < 8; ++j)
#pragma unroll
    for (int v = 0; v < 8; ++v)
      out[(m0 + v + 8 * half) * CC2 + j * 16 + lm] = acc[j][v];
}

// ---------------------------------------------------------------------------
// Stage 3: out[b][n][cb] = sum_ca q[b][ca][n] * KV[b][ca][cb]
// KV (summed over the 8 K-slice partials) staged pair-interleaved in LDS;
// output written row-major [N][C2] -- the torch reshape then reinterprets it
// as [C2][N] for stage 4 (no data movement).
// ---------------------------------------------------------------------------
__global__ __launch_bounds__(256) void attn_out_kernel(
    const float* __restrict__ qkv, const float* __restrict__ kvpart,
    float* __restrict__ outflat) {
  __shared__ float kvs[(KCH / 2) * PS];
  const int b    = blockIdx.y;
  const int wave = threadIdx.x >> 5;
  const int lane = threadIdx.x & 31;
  const int half = lane >> 4, lm = lane & 15;
  const int n0   = blockIdx.x * 128 + wave * 16;  // N-strip per wave

  const float* qbuf = qkv + ((size_t)b * CC2) * NN;

  v8f acc[8];
#pragma unroll
  for (int j = 0; j < 8; ++j) acc[j] = vzero();

  for (int kc = 0; kc < CC2; kc += KCH) {
    __syncthreads();
    for (int idx = threadIdx.x; idx < (KCH / 2) * 128; idx += 256) {
      const int pr = idx >> 7, cc = idx & 127;
      const int r0 = kc + 2 * pr;
      float s0 = 0.f, s1 = 0.f;
#pragma unroll
      for (int sl = 0; sl < 8; ++sl) {
        const float* kp = kvpart + ((size_t)(sl * 4 + b) * CC2 + r0) * CC2 + cc;
        s0 += kp[0];
        s1 += kp[CC2];
      }
      v2f sv = {s0, s1};
      *reinterpret_cast<v2f*>(&kvs[pr * PS + cc * 2]) = sv;
    }
    __syncthreads();
#pragma unroll 4
    for (int kk = 0; kk < KCH; kk += 4) {
      const int k = kc + kk + 2 * half;
      v2f a = {qbuf[(size_t)k * NN + n0 + lm],
               qbuf[(size_t)(k + 1) * NN + n0 + lm]};
      const float* bbase = &kvs[((kk >> 1) + half) * PS + lm * 2];
#pragma unroll
      for (int j = 0; j < 8; ++j) {
        const v2f bf = *reinterpret_cast<const v2f*>(bbase + j * 32);
        acc[j] = wmma4(a, bf, acc[j]);
      }
    }
  }

  float* ob = outflat + (size_t)b * NN * CC2;
#pragma unroll
  for (int j = 0; j < 8; ++j)
#pragma unroll
    for (int v = 0; v < 8; ++v)
      ob[(size_t)(n0 + v + 8 * half) * CC2 + j * 16 + lm] = acc[j][v];
}

// ---------------------------------------------------------------------------
// Stage 4: result[b][o][n] = sum_c Wo[o][c] * out_view[b][c][n] + X[b][o][n]
// out_view = stage-3 buffer reinterpreted row-major [C2][N] (torch reshape).
// Block: 512 thr (16 waves = 16 M-strips covering M=256); the 128-wide
// out_view slab is staged pair-interleaved in LDS, reused by all 16 M-strips.
// ---------------------------------------------------------------------------
__global__ __launch_bounds__(512) void out_conv_kernel(
    const float* __restrict__ outflat, const float* __restrict__ Wo,
    const float* __restrict__ X, float* __restrict__ out) {
  __shared__ float bs[(KCH / 2) * PS];
  const int b    = blockIdx.y;
  const int n0   = blockIdx.x * 128;
  const int wave = threadIdx.x >> 5;  // 0..15 -> M-strip
  const int lane = threadIdx.x & 31;
  const int half = lane >> 4, lm = lane & 15;
  const int m0   = wave * 16;

  const float* ov = outflat + (size_t)b * NN * CC2;  // view as [CC2][NN]

  v8f acc[8];
#pragma unroll
  for (int j = 0; j < 8; ++j) acc[j] = vzero();

  for (int kc = 0; kc < CC2; kc += KCH) {
    __syncthreads();
    for (int idx = threadIdx.x; idx < (KCH / 2) * 128; idx += 512) {
      const int pr = idx >> 7, cc = idx & 127;
      const int r0 = kc + 2 * pr;
      v2f bv = {ov[(size_t)r0 * NN + n0 + cc], ov[(size_t)(r0 + 1) * NN + n0 + cc]};
      *reinterpret_cast<v2f*>(&bs[pr * PS + cc * 2]) = bv;
    }
    __syncthreads();
#pragma unroll 4
    for (int kk = 0; kk < KCH; kk += 4) {
      const v2f a =
          *reinterpret_cast<const v2f*>(Wo + (m0 + lm) * CC2 + kc + kk + 2 * half);
      const float* bbase = &bs[((kk >> 1) + half) * PS + lm * 2];
#pragma unroll
      for (int j = 0; j < 8; ++j) {
        const v2f bf = *reinterpret_cast<const v2f*>(bbase + j * 32);
        acc[j] = wmma4(a, bf, acc[j]);
      }
    }
  }

#pragma unroll
  for (int j = 0; j < 8; ++j)
#pragma unroll
    for (int v = 0; v < 8; ++v) {
      const int    o  = m0 + v + 8 * half;
      const int    n  = n0 + j * 16 + lm;
      const size_t gi = ((size_t)b * CC + o) * NN + n;
      out[gi] = acc[j][v] + X[gi];
    }
}

// ---------------------------------------------------------------------------
extern "C" void kernel_launch(void* const* d_in, const int* in_sizes, int n_in,
                              void* d_out, int out_size, void* d_ws,
                              size_t ws_size, hipStream_t stream) {
  (void)in_sizes; (void)n_in; (void)out_size; (void)ws_size;
  const float* X  = (const float*)d_in[0];
  const float* pe = (const float*)d_in[1];
  const float* tq = (const float*)d_in[2];
  const float* tk = (const float*)d_in[3];
  const float* tv = (const float*)d_in[4];
  const float* Wq = (const float*)d_in[5];
  const float* bq = (const float*)d_in[6];
  const float* Wk = (const float*)d_in[7];
  const float* bk = (const float*)d_in[8];
  const float* Wv = (const float*)d_in[9];
  const float* bv = (const float*)d_in[10];
  const float* Wo = (const float*)d_in[11];

  float* ws      = (float*)d_ws;
  float* qkv     = ws;                                   // 3*4*128*4096 = 6,291,456 f
  float* kvpart  = qkv + (size_t)3 * 4 * CC2 * NN;       // 8*4*128*128  =   524,288 f
  float* outflat = kvpart + (size_t)8 * 4 * CC2 * CC2;   // 4*4096*128   = 2,097,152 f
                                                         // total ~35.7 MB of d_ws

  proj_kernel<<<dim3(32, 4), 768, 0, stream>>>(X, pe, tq, tk, tv, Wq, bq, Wk, bk,
                                               Wv, bv, qkv);
  kv_kernel<<<dim3(8, 4), 256, 0, stream>>>(qkv, kvpart);
  attn_out_kernel<<<dim3(32, 4), 256, 0, stream>>>(qkv, kvpart, outflat);
  out_conv_kernel<<<dim3(32, 4), 512, 0, stream>>>(outflat, Wo, X, (float*)d_out);
}